// Net_17394617549527
// MI455X (gfx1250) — compile-verified
//
#include <hip/hip_runtime.h>
#include <math.h>

// ---------------------------------------------------------------------------
// CDNA5 WMMA f32 16x16x4:  A = 16x4 f32 (2 VGPRs), B = 4x16 f32 (2 VGPRs),
// C/D = 16x16 f32 (8 VGPRs).  wave32.
// ---------------------------------------------------------------------------
typedef __attribute__((ext_vector_type(2))) float v2f;
typedef __attribute__((ext_vector_type(8))) float v8f;

#define S_MUL 64
#define V_MUL 32
#define EPS   1e-5f

// ---------------------------------------------------------------------------
// zero fill
// ---------------------------------------------------------------------------
__global__ void zero_kernel(float* __restrict__ p, long n) {
    long t = (long)blockIdx.x * blockDim.x + threadIdx.x;
    if (t < n) p[t] = 0.0f;
}

// ---------------------------------------------------------------------------
// Node pre-pass: h_s = node_s@W_pre_s/8 + b,  h_v = node_v@W_pre_v/sqrt(32),
// sc_s = node_s @ W_sc_s[:,spec,:]/sqrt(128), sc_v analog /sqrt(64).
// blockDim = 64, one block per node.
// ---------------------------------------------------------------------------
__global__ void node_pre_kernel(const float* __restrict__ node_s,
                                const float* __restrict__ node_v,
                                const float* __restrict__ one_hot,
                                const float* __restrict__ W_pre_s,
                                const float* __restrict__ b_pre_s,
                                const float* __restrict__ W_pre_v,
                                const float* __restrict__ W_sc_s,
                                const float* __restrict__ W_sc_v,
                                float* __restrict__ h_s,
                                float* __restrict__ h_v,
                                float* __restrict__ sc_s,
                                float* __restrict__ sc_v) {
    int n = blockIdx.x;
    int o = threadIdx.x;           // 0..63
    float oh0 = one_hot[n * 2 + 0];
    float oh1 = one_hot[n * 2 + 1];

    float acc = 0.0f, accsc = 0.0f;
    for (int u = 0; u < S_MUL; ++u) {
        float x = node_s[(size_t)n * S_MUL + u];
        acc   += x * W_pre_s[u * S_MUL + o];
        accsc += x * (oh0 * W_sc_s[(u * 2 + 0) * S_MUL + o] +
                      oh1 * W_sc_s[(u * 2 + 1) * S_MUL + o]);
    }
    h_s[(size_t)n * S_MUL + o]  = acc * 0.125f + b_pre_s[o];          // /sqrt(64)
    sc_s[(size_t)n * S_MUL + o] = accsc * 0.08838834764831845f;       // /sqrt(128)

    if (o < V_MUL) {
        float a0 = 0, a1 = 0, a2 = 0, s0 = 0, s1 = 0, s2 = 0;
        for (int u = 0; u < V_MUL; ++u) {
            float wv  = W_pre_v[u * V_MUL + o];
            float wsc = oh0 * W_sc_v[(u * 2 + 0) * V_MUL + o] +
                        oh1 * W_sc_v[(u * 2 + 1) * V_MUL + o];
            size_t b = ((size_t)n * V_MUL + u) * 3;
            float x0 = node_v[b + 0], x1 = node_v[b + 1], x2 = node_v[b + 2];
            a0 += x0 * wv;  a1 += x1 * wv;  a2 += x2 * wv;
            s0 += x0 * wsc; s1 += x1 * wsc; s2 += x2 * wsc;
        }
        const float invv = 0.17677669529663687f;  // 1/sqrt(32)
        size_t b = ((size_t)n * V_MUL + o) * 3;
        h_v[b + 0] = a0 * invv;  h_v[b + 1] = a1 * invv;  h_v[b + 2] = a2 * invv;
        sc_v[b + 0] = s0 * 0.125f; sc_v[b + 1] = s1 * 0.125f; sc_v[b + 2] = s2 * 0.125f; // /sqrt(64)
    }
}

// ---------------------------------------------------------------------------
// Gather s_in = [h_s[i] | h_s[j] | edge_s]  (E x 192)
// ---------------------------------------------------------------------------
__global__ void gather_s_kernel(const float* __restrict__ h_s,
                                const float* __restrict__ edge_s,
                                const int* __restrict__ ei, int E,
                                float* __restrict__ s_in) {
    long t = (long)blockIdx.x * blockDim.x + threadIdx.x;
    if (t >= (long)E * 192) return;
    int e = (int)(t / 192);
    int u = (int)(t % 192);
    float v;
    if (u < 64) {
        int i = ei[e];
        v = h_s[(size_t)i * 64 + u];
    } else if (u < 128) {
        int j = ei[E + e];
        v = h_s[(size_t)j * 64 + (u - 64)];
    } else {
        v = edge_s[(size_t)e * 64 + (u - 128)];
    }
    s_in[t] = v;
}

// ---------------------------------------------------------------------------
// Gather v_in component planes (3 x E x 96) and vdot = v_in . Y1  (E x 96)
// ---------------------------------------------------------------------------
__global__ void gather_v_kernel(const float* __restrict__ h_v,
                                const float* __restrict__ edge_v,
                                const float* __restrict__ edge_sh,
                                const int* __restrict__ ei, int E,
                                float* __restrict__ vc0, float* __restrict__ vc1,
                                float* __restrict__ vc2, float* __restrict__ vdot) {
    long t = (long)blockIdx.x * blockDim.x + threadIdx.x;
    if (t >= (long)E * 96) return;
    int e = (int)(t / 96);
    int u = (int)(t % 96);
    const float* src;
    size_t base;
    if (u < 32) {
        src = h_v;  base = ((size_t)ei[e] * 32 + u) * 3;
    } else if (u < 64) {
        src = h_v;  base = ((size_t)ei[E + e] * 32 + (u - 32)) * 3;
    } else {
        src = edge_v; base = ((size_t)e * 32 + (u - 64)) * 3;
    }
    float x0 = src[base + 0], x1 = src[base + 1], x2 = src[base + 2];
    vc0[t] = x0; vc1[t] = x1; vc2[t] = x2;
    vdot[t] = x0 * edge_sh[e * 4 + 1] + x1 * edge_sh[e * 4 + 2] + x2 * edge_sh[e * 4 + 3];
}

// ---------------------------------------------------------------------------
// WMMA GEMM with async-staged LDS weights:
//   C[MxN] = act(scale * A[MxK] @ B[KxN] + bias[N]),  K,N compile-time,
//   N multiple of 16, K multiple of 4, M multiple of 16.
//
// Block = 256 threads (8 waves).  The whole B matrix (<= 73.7 KB) is staged
// into LDS once per block via GLOBAL_LOAD_ASYNC_TO_LDS_B128 (ASYNCcnt path),
// then each wave computes a full 16-row x N strip: one A fragment per k-step
// feeds N/16 WMMAs, so the streaming A operand is read exactly once.
//
// A-frag layout (ISA 16x4): row = lane&15 for all lanes; VGPR0/1 hold
// K = (0,1) for lanes 0-15 and K = (2,3) for lanes 16-31.
// B-frag (4x16, row striped across lanes): K row khalf / khalf+1, N = lane&15.
// C/D: VGPR r -> rows r (lanes 0-15) and r+8 (lanes 16-31), N = lane&15.
// ---------------------------------------------------------------------------
template <int K, int N>
__global__ __launch_bounds__(256) void gemm_wmma(
        const float* __restrict__ A, const float* __restrict__ B,
        float* __restrict__ C, int M,
        float scale, const float* __restrict__ bias, int act) {
    constexpr int TN = N / 16;
    __shared__ float ldsB[K * N];

    int tid = threadIdx.x;

    // ---- stage B into LDS with async b128 copies (ASYNCcnt) ----
    for (int idx = tid * 4; idx < K * N; idx += 256 * 4) {
        unsigned lds_off = (unsigned)(uintptr_t)(&ldsB[idx]);
        const float* gp = B + idx;
        asm volatile("global_load_async_to_lds_b128 %0, %1, off"
                     :: "v"(lds_off), "v"(gp)
                     : "memory");
    }
    asm volatile("s_wait_asynccnt 0x0" ::: "memory");
    __syncthreads();

    int wave = tid >> 5;
    int lane = tid & 31;
    int tm = blockIdx.x * 8 + wave;          // 16-row strip per wave
    if (tm * 16 >= M) return;                // wave-uniform: EXEC all-ones at WMMA

    int row   = tm * 16 + (lane & 15);
    int colh  = lane & 15;
    int khalf = (lane >> 4) << 1;            // 0 for lanes 0-15, 2 for 16-31

    v8f acc[TN];
    v8f zero = {};
#pragma unroll
    for (int t = 0; t < TN; ++t) acc[t] = zero;

    const float* arow = A + (size_t)row * K + khalf;
    for (int k = 0; k < K; k += 4) {
        v2f a;
        a[0] = arow[k];
        a[1] = arow[k + 1];
#pragma unroll
        for (int t = 0; t < TN; ++t) {
            v2f b;
            b[0] = ldsB[(k + khalf) * N + t * 16 + colh];
            b[1] = ldsB[(k + khalf + 1) * N + t * 16 + colh];
            acc[t] = __builtin_amdgcn_wmma_f32_16x16x4_f32(
                    false, a, false, b, (short)0, acc[t], false, false);
        }
    }

    int rtop = (lane >> 4) << 3;             // +8 row offset for upper half lanes
#pragma unroll
    for (int t = 0; t < TN; ++t) {
        int col = t * 16 + colh;
        float bc = bias ? bias[col] : 0.0f;
#pragma unroll
        for (int r = 0; r < 8; ++r) {
            float v = acc[t][r] * scale + bc;
            if (act == 1) v = v / (1.0f + __expf(-v));   // SiLU
            C[(size_t)(tm * 16 + rtop + r) * N + col] = v;
        }
    }
}

// ---------------------------------------------------------------------------
// Per-edge combine + scatter (segment sum via f32 atomics).
// blockDim = 96, one block per edge.
// ---------------------------------------------------------------------------
__global__ void edge_combine_scatter(const float* __restrict__ zs0,
                                     const float* __restrict__ zs1,
                                     const float* __restrict__ zsv,
                                     const float* __restrict__ zv0,
                                     const float* __restrict__ zv1,
                                     const float* __restrict__ zv2,
                                     const float* __restrict__ w,
                                     const float* __restrict__ edge_sh,
                                     const int* __restrict__ ei, int E,
                                     float* __restrict__ agg_s,
                                     float* __restrict__ agg_v) {
    int e = blockIdx.x;
    int t = threadIdx.x;           // 0..95
    int i = ei[e];
    const float inv2 = 0.7071067811865476f;
    float Y0 = edge_sh[e * 4 + 0];
    float zs = (zs0[(size_t)e * 96 + t] * Y0 + zs1[(size_t)e * 96 + t]) * inv2;
    float sig = 1.0f / (1.0f + __expf(-zs));
    if (t < 64) {
        float os = zs * sig * w[(size_t)e * 96 + t];
        atomicAdd(&agg_s[(size_t)i * 64 + t], os);
    } else {
        int o = t - 64;
        float gw = sig * w[(size_t)e * 96 + 64 + o];
        float sv = zsv[(size_t)e * 32 + o];
        float y1x = edge_sh[e * 4 + 1], y1y = edge_sh[e * 4 + 2], y1z = edge_sh[e * 4 + 3];
        float zx = (zv0[(size_t)e * 32 + o] * Y0 + sv * y1x) * inv2;
        float zy = (zv1[(size_t)e * 32 + o] * Y0 + sv * y1y) * inv2;
        float zz = (zv2[(size_t)e * 32 + o] * Y0 + sv * y1z) * inv2;
        size_t b = ((size_t)i * 32 + o) * 3;
        atomicAdd(&agg_v[b + 0], gw * zx);
        atomicAdd(&agg_v[b + 1], gw * zy);
        atomicAdd(&agg_v[b + 2], gw * zz);
    }
}

// ---------------------------------------------------------------------------
// Node post-pass: post matmuls + sc residual, layernorm (s), vector norm (v),
// residual with node_s/node_v, write (N,160) output.  blockDim = 64.
// ---------------------------------------------------------------------------
__global__ void node_post_kernel(const float* __restrict__ node_s,
                                 const float* __restrict__ node_v,
                                 const float* __restrict__ agg_s,
                                 const float* __restrict__ agg_v,
                                 const float* __restrict__ W_post_s,
                                 const float* __restrict__ b_post_s,
                                 const float* __restrict__ W_post_v,
                                 const float* __restrict__ sc_s,
                                 const float* __restrict__ sc_v,
                                 const float* __restrict__ gamma_s,
                                 const float* __restrict__ beta_s,
                                 const float* __restrict__ gamma_v,
                                 float* __restrict__ out) {
    __shared__ float red[64];
    int n = blockIdx.x;
    int o = threadIdx.x;           // 0..63

    float acc = 0.0f;
    for (int u = 0; u < 64; ++u)
        acc += agg_s[(size_t)n * 64 + u] * W_post_s[u * 64 + o];
    float as = acc * 0.125f + b_post_s[o] + sc_s[(size_t)n * 64 + o];

    // mean
    red[o] = as; __syncthreads();
    for (int s = 32; s > 0; s >>= 1) { if (o < s) red[o] += red[o + s]; __syncthreads(); }
    float mean = red[0] * (1.0f / 64.0f); __syncthreads();
    float cs = as - mean;
    // var
    red[o] = cs * cs; __syncthreads();
    for (int s = 32; s > 0; s >>= 1) { if (o < s) red[o] += red[o + s]; __syncthreads(); }
    float var = red[0] * (1.0f / 64.0f); __syncthreads();
    float ns = cs * rsqrtf(var + EPS) * gamma_s[o] + beta_s[o];
    out[(size_t)n * 160 + o] = node_s[(size_t)n * 64 + o] + ns;

    // vector channel
    float av0 = 0, av1 = 0, av2 = 0;
    if (o < 32) {
        for (int u = 0; u < 32; ++u) {
            float wv = W_post_v[u * 32 + o];
            size_t b = ((size_t)n * 32 + u) * 3;
            av0 += agg_v[b + 0] * wv; av1 += agg_v[b + 1] * wv; av2 += agg_v[b + 2] * wv;
        }
        const float inv = 0.17677669529663687f;  // 1/sqrt(32)
        size_t b = ((size_t)n * 32 + o) * 3;
        av0 = av0 * inv + sc_v[b + 0];
        av1 = av1 * inv + sc_v[b + 1];
        av2 = av2 * inv + sc_v[b + 2];
    }
    red[o] = (o < 32) ? (av0 * av0 + av1 * av1 + av2 * av2) : 0.0f; __syncthreads();
    for (int s = 32; s > 0; s >>= 1) { if (o < s) red[o] += red[o + s]; __syncthreads(); }
    float vnorm = red[0] * (1.0f / 32.0f);
    if (o < 32) {
        float g = gamma_v[o] * rsqrtf(vnorm + EPS);
        size_t b = ((size_t)n * 32 + o) * 3;
        out[(size_t)n * 160 + 64 + o * 3 + 0] = node_v[b + 0] + av0 * g;
        out[(size_t)n * 160 + 64 + o * 3 + 1] = node_v[b + 1] + av1 * g;
        out[(size_t)n * 160 + 64 + o * 3 + 2] = node_v[b + 2] + av2 * g;
    }
}

// ---------------------------------------------------------------------------
// host side
// ---------------------------------------------------------------------------
static inline int gemm_blocks(int M) { return ((M / 16) + 7) / 8; }

extern "C" void kernel_launch(void* const* d_in, const int* in_sizes, int n_in,
                              void* d_out, int out_size, void* d_ws, size_t ws_size,
                              hipStream_t stream) {
    const float* node_s   = (const float*)d_in[0];
    const float* node_v   = (const float*)d_in[1];
    const float* edge_s   = (const float*)d_in[2];
    const float* edge_v   = (const float*)d_in[3];
    const float* edge_sh  = (const float*)d_in[4];
    const float* edge_len = (const float*)d_in[5];
    const float* one_hot  = (const float*)d_in[6];
    const int*   eidx     = (const int*)d_in[7];
    // d_in[8] = batch (unused)
    const float* W_pre_s  = (const float*)d_in[9];
    const float* b_pre_s  = (const float*)d_in[10];
    const float* W_pre_v  = (const float*)d_in[11];
    const float* W_tp_s0  = (const float*)d_in[12];
    const float* W_tp_v1  = (const float*)d_in[13];
    const float* W_tp_v0  = (const float*)d_in[14];
    const float* W_tp_s1  = (const float*)d_in[15];
    const float* fc_w1    = (const float*)d_in[16];
    const float* fc_b1    = (const float*)d_in[17];
    const float* fc_w2    = (const float*)d_in[18];
    const float* fc_b2    = (const float*)d_in[19];
    const float* fc_w3    = (const float*)d_in[20];
    const float* fc_b3    = (const float*)d_in[21];
    const float* W_post_s = (const float*)d_in[22];
    const float* b_post_s = (const float*)d_in[23];
    const float* W_post_v = (const float*)d_in[24];
    const float* W_sc_s   = (const float*)d_in[25];
    const float* W_sc_v   = (const float*)d_in[26];
    const float* gamma_s  = (const float*)d_in[27];
    const float* beta_s   = (const float*)d_in[28];
    const float* gamma_v  = (const float*)d_in[29];

    const int N = in_sizes[0] / 64;   // nodes
    const int E = in_sizes[2] / 64;   // edges

    // workspace carve-out (floats)
    float* ws = (float*)d_ws;
    size_t off = 0;
    auto take = [&](size_t cnt) { float* p = ws + off; off += cnt; return p; };
    float* h_s   = take((size_t)N * 64);
    float* h_v   = take((size_t)N * 96);
    float* sc_s  = take((size_t)N * 64);
    float* sc_v  = take((size_t)N * 96);
    float* agg_s = take((size_t)N * 64);   // agg_s + agg_v contiguous: one zero pass
    float* agg_v = take((size_t)N * 96);
    float* s_in  = take((size_t)E * 192);
    float* vc0   = take((size_t)E * 96);
    float* vc1   = take((size_t)E * 96);
    float* vc2   = take((size_t)E * 96);
    float* vdot  = take((size_t)E * 96);
    float* zs0   = take((size_t)E * 96);
    float* zs1   = take((size_t)E * 96);
    float* zsv   = take((size_t)E * 32);
    float* zv0   = take((size_t)E * 32);
    float* zv1   = take((size_t)E * 32);
    float* zv2   = take((size_t)E * 32);
    float* fc1   = take((size_t)E * 64);
    float* fc2   = take((size_t)E * 64);
    float* wbuf  = take((size_t)E * 96);
    (void)ws_size;

    const float rs192 = 1.0f / sqrtf(192.0f);
    const float rs96  = 1.0f / sqrtf(96.0f);
    const float dot96 = (1.0f / sqrtf(3.0f)) / sqrtf(96.0f);

    // 0) zero the aggregation buffers (atomics accumulate into them)
    {
        long n = (long)N * 160;
        zero_kernel<<<(int)((n + 255) / 256), 256, 0, stream>>>(agg_s, n);
    }
    // 1) node pre-pass
    node_pre_kernel<<<N, 64, 0, stream>>>(node_s, node_v, one_hot, W_pre_s, b_pre_s,
                                          W_pre_v, W_sc_s, W_sc_v,
                                          h_s, h_v, sc_s, sc_v);
    // 2) gathers
    {
        long n = (long)E * 192;
        gather_s_kernel<<<(int)((n + 255) / 256), 256, 0, stream>>>(h_s, edge_s, eidx, E, s_in);
        n = (long)E * 96;
        gather_v_kernel<<<(int)((n + 255) / 256), 256, 0, stream>>>(h_v, edge_v, edge_sh, eidx, E,
                                                                    vc0, vc1, vc2, vdot);
    }
    // 3) WMMA GEMMs (f32 16x16x4, B async-staged in LDS)
    int gb = gemm_blocks(E);
    gemm_wmma<192, 96><<<gb, 256, 0, stream>>>(s_in, W_tp_s0, zs0, E, rs192, nullptr, 0);
    gemm_wmma< 96, 96><<<gb, 256, 0, stream>>>(vdot, W_tp_v1, zs1, E, dot96, nullptr, 0);
    gemm_wmma<192, 32><<<gb, 256, 0, stream>>>(s_in, W_tp_s1, zsv, E, rs192, nullptr, 0);
    gemm_wmma< 96, 32><<<gb, 256, 0, stream>>>(vc0,  W_tp_v0, zv0, E, rs96,  nullptr, 0);
    gemm_wmma< 96, 32><<<gb, 256, 0, stream>>>(vc1,  W_tp_v0, zv1, E, rs96,  nullptr, 0);
    gemm_wmma< 96, 32><<<gb, 256, 0, stream>>>(vc2,  W_tp_v0, zv2, E, rs96,  nullptr, 0);
    gemm_wmma<128, 64><<<gb, 256, 0, stream>>>(edge_len, fc_w1, fc1, E, 1.0f, fc_b1, 1); // SiLU
    gemm_wmma< 64, 64><<<gb, 256, 0, stream>>>(fc1,      fc_w2, fc2, E, 1.0f, fc_b2, 1); // SiLU
    gemm_wmma< 64, 96><<<gb, 256, 0, stream>>>(fc2,      fc_w3, wbuf, E, 1.0f, fc_b3, 0);
    // 4) combine + scatter-add
    edge_combine_scatter<<<E, 96, 0, stream>>>(zs0, zs1, zsv, zv0, zv1, zv2, wbuf,
                                               edge_sh, eidx, E, agg_s, agg_v);
    // 5) node post-pass -> output (N,160)
    node_post_kernel<<<N, 64, 0, stream>>>(node_s, node_v, agg_s, agg_v,
                                           W_post_s, b_post_s, W_post_v,
                                           sc_s, sc_v, gamma_s, beta_s, gamma_v,
                                           (float*)d_out);
    (void)out_size; (void)n_in;
}